// DDApprox_18631568130862
// MI455X (gfx1250) — compile-verified
//
#include <hip/hip_runtime.h>

// DDOpt (Wilson-Dirac D^dagger D on 8x8 lattice, 2 spins) for MI455X / gfx1250.
//
// Per gauge config b, D_b is a dense 128x128 complex operator; using
// gamma5-hermiticity:  out = g5 . D_b ( g5 . (D_b psi) ).
// One block per config: build D_b in LDS once, then apply it twice to tiles of
// 16 basis vectors with V_WMMA_F32_16X16X4_F32 (full fp32 precision).
// Basis tiles are double-buffered with GLOBAL_LOAD_ASYNC_TO_LDS_B128
// (ASYNCcnt-tracked direct-to-LDS DMA) so global staging overlaps the WMMA
// stages.  Output (128 MiB complex64) is written exactly once -> near the
// 23.3 TB/s HBM roofline floor (~6 us of traffic).

typedef float v2f __attribute__((ext_vector_type(2)));
typedef float v8f __attribute__((ext_vector_type(8)));

#define NC    128          // spinor components per field (8*8*2)
#define NK    512          // basis vectors
#define NB    256          // gauge configs
#define KAPPA 0.276f
#define DSTR  132          // padded LDS row stride (floats): 132 % 64 == 4 -> conflict-free frags
#define KTILE 16           // basis vectors per tile (one WMMA N-tile)
#define NTILES (NK / KTILE)

// Async-stage one 16x128-float tile (contiguous 8 KB in global) into
// DSTR-padded LDS rows.  Each wave issues 2 x B128 (32 lanes x 16 B = 512 B
// per instruction); 8 waves cover the tile.  ldsFloatBase = float offset of
// the destination buffer from the start of dynamic LDS.
__device__ __forceinline__ void async_load_tile(const float* g, int ldsFloatBase,
                                                int wv, int lane)
{
#pragma unroll
    for (int q = 0; q < 2; ++q) {
        const int e       = (wv * 2 + q) * 32 + lane;          // 16-byte chunk id
        const int kk      = e >> 5;                            // tile row (k)
        const int j4      = (e & 31) * 4;                      // column (j)
        const int ldsoff  = (ldsFloatBase + kk * DSTR + j4) * 4;
        const unsigned long long ga = (unsigned long long)(uintptr_t)(g + e * 4);
        asm volatile("global_load_async_to_lds_b128 %0, %1, off"
                     :: "v"(ldsoff), "v"(ga) : "memory");
    }
}

// One 16x16 tile of a complex GEMM X = A*B over the full K=128 reduction.
// A fragments from rows [arow] of (Are,Aim); B fragments from rows [brow] of
// (Bre,Bim) stored transposed ([n][j], stride DSTR).  coff = 2*(lane>=16).
__device__ __forceinline__ void tile_cgemm128(
    const float* Are, const float* Aim,
    const float* Bre, const float* Bim,
    int arow, int brow, int coff,
    v8f& outRe, v8f& outIm)
{
    v8f cre = {0.f,0.f,0.f,0.f,0.f,0.f,0.f,0.f};
    v8f cim = {0.f,0.f,0.f,0.f,0.f,0.f,0.f,0.f};
    const float* ar_base = Are + (size_t)arow * DSTR;
    const float* ai_base = Aim + (size_t)arow * DSTR;
    const float* br_base = Bre + (size_t)brow * DSTR;
    const float* bi_base = Bim + (size_t)brow * DSTR;
#pragma unroll 4
    for (int jt = 0; jt < NC / 4; ++jt) {
        const int col = jt * 4 + coff;
        v2f ar = *(const v2f*)(ar_base + col);
        v2f ai = *(const v2f*)(ai_base + col);
        v2f br = *(const v2f*)(br_base + col);
        v2f bi = *(const v2f*)(bi_base + col);
        v2f ain = -ai;   // Xre -= Aim*Bim (f32 WMMA has no A-negate modifier)
        // 8-arg form: (neg_a, A, neg_b, B, c_mod, C, reuse_a, reuse_b)
        cre = __builtin_amdgcn_wmma_f32_16x16x4_f32(false, ar,  false, br, (short)0, cre, false, false);
        cre = __builtin_amdgcn_wmma_f32_16x16x4_f32(false, ain, false, bi, (short)0, cre, false, false);
        cim = __builtin_amdgcn_wmma_f32_16x16x4_f32(false, ar,  false, bi, (short)0, cim, false, false);
        cim = __builtin_amdgcn_wmma_f32_16x16x4_f32(false, ai,  false, br, (short)0, cim, false, false);
    }
    outRe = cre; outIm = cim;
}

__global__ void __launch_bounds__(256) dd_wilson_wmma(
    const float* __restrict__ u_re, const float* __restrict__ u_im,
    const float* __restrict__ b_re, const float* __restrict__ b_im,
    float* __restrict__ out)
{
    extern __shared__ float smem[];
    float* Dre = smem;                         // 128 x DSTR
    float* Dim = Dre + NC * DSTR;              // 128 x DSTR
    float* P   = Dim + NC * DSTR;              // 2 buffers x (re,im) x KTILE x DSTR
    float* Tre = P + 4 * KTILE * DSTR;         // KTILE x DSTR (g5 * D psi, [k][i])
    float* Tim = Tre + KTILE * DSTR;
    const int PBASE = 2 * NC * DSTR;           // float offset of P from LDS start

    const int t    = threadIdx.x;
    const int b    = blockIdx.x;
    const int lane = t & 31;
    const int ln   = lane & 15;                // N index / A-fragment M row
    const int half = lane >> 4;                // 0: K 0,1 / M 0..7 ; 1: K 2,3 / M 8..15
    const int m    = t >> 5;                   // wave id == M-tile (8 x 16 rows = 128)
    const int coff = half * 2;

    // ---- kick off async stage of basis tile 0 (overlaps D build) ----
    async_load_tile(b_re, PBASE,                 m, lane);
    async_load_tile(b_im, PBASE + KTILE * DSTR,  m, lane);

    // ---- zero D (including padding) ----
    for (int idx = t; idx < NC * DSTR; idx += 256) { Dre[idx] = 0.f; Dim[idx] = 0.f; }
    __syncthreads();

    // ---- build dense Wilson-Dirac matrix D_b: 9 nonzeros per row ----
    if (t < NC) {
        const int i = t, s = i & 1, site = i >> 1, y = site & 7, x = site >> 3;
        const float mk = -KAPPA;
        const float* Ur = u_re + (size_t)b * 128;   // [mu][x][y]
        const float* Ui = u_im + (size_t)b * 128;

        auto put = [&](int col, float pr, float pi, float ur, float ui) {
            Dre[i * DSTR + col] = mk * (pr * ur - pi * ui);
            Dim[i * DSTR + col] = mk * (pr * ui + pi * ur);
        };

        Dre[i * DSTR + i] = 1.f;                    // identity term

        const int xp = (x + 1) & 7, xm = (x - 1) & 7;
        const int yp = (y + 1) & 7, ym = (y - 1) & 7;

        const float u0r  = Ur[x * 8 + y],        u0i  = Ui[x * 8 + y];
        const float u0br = Ur[xm * 8 + y],       u0bi = -Ui[xm * 8 + y];       // conj
        const float u1r  = Ur[64 + x * 8 + y],   u1i  = Ui[64 + x * 8 + y];
        const float u1br = Ur[64 + x * 8 + ym],  u1bi = -Ui[64 + x * 8 + ym];  // conj

        // mu=0 forward:  (1-g0) = [[1,-1],[-1,1]]
        const int cf0 = (xp * 8 + y) * 2;
        put(cf0 + s,        1.f, 0.f, u0r, u0i);
        put(cf0 + (1 - s), -1.f, 0.f, u0r, u0i);
        // mu=0 backward: (1+g0) = [[1,1],[1,1]]
        const int cb0 = (xm * 8 + y) * 2;
        put(cb0 + 0, 1.f, 0.f, u0br, u0bi);
        put(cb0 + 1, 1.f, 0.f, u0br, u0bi);
        // mu=1 forward:  (1-g1) = [[1,+i],[-i,1]]
        const int cf1 = (x * 8 + yp) * 2;
        put(cf1 + s,       1.f, 0.f,                   u1r, u1i);
        put(cf1 + (1 - s), 0.f, (s == 0) ? 1.f : -1.f, u1r, u1i);
        // mu=1 backward: (1+g1) = [[1,-i],[+i,1]]
        const int cb1 = (x * 8 + ym) * 2;
        put(cb1 + s,       1.f, 0.f,                   u1br, u1bi);
        put(cb1 + (1 - s), 0.f, (s == 0) ? -1.f : 1.f, u1br, u1bi);
    }

    // ---- stream tiles of 16 basis vectors (double-buffered async staging) ----
    for (int kt = 0; kt < NTILES; ++kt) {
        const int k0  = kt * KTILE;
        const int sel = kt & 1;
        const float* PreC = P + sel * 2 * KTILE * DSTR;
        const float* PimC = PreC + KTILE * DSTR;

        // my async loads into the current buffer are done; barrier makes every
        // wave's chunks (and, on kt==0, the D matrix) visible block-wide
        asm volatile("s_wait_asynccnt 0" ::: "memory");
        __syncthreads();

        // issue DMA for the next tile into the other buffer; it fills while we
        // run both WMMA stages below
        if (kt + 1 < NTILES) {
            const int nb = PBASE + (1 - sel) * 2 * KTILE * DSTR;
            async_load_tile(b_re + (size_t)(k0 + KTILE) * NC, nb,              m, lane);
            async_load_tile(b_im + (size_t)(k0 + KTILE) * NC, nb + KTILE*DSTR, m, lane);
        }

        // stage 1: X = D * Psi ; store T = g5 . X  (row parity == v parity)
        v8f xre, xim;
        tile_cgemm128(Dre, Dim, PreC, PimC, m * 16 + ln, ln, coff, xre, xim);
#pragma unroll
        for (int v = 0; v < 8; ++v) {
            const float s5 = (v & 1) ? -1.f : 1.f;
            const int i = m * 16 + half * 8 + v;
            Tre[ln * DSTR + i] = s5 * xre[v];
            Tim[ln * DSTR + i] = s5 * xim[v];
        }
        __syncthreads();

        // stage 2: Y = D * T ; out = g5 . Y  -> interleaved complex64
        tile_cgemm128(Dre, Dim, Tre, Tim, m * 16 + ln, ln, coff, xre, xim);
#pragma unroll
        for (int v = 0; v < 8; ++v) {
            const float s5 = (v & 1) ? -1.f : 1.f;
            const int i = m * 16 + half * 8 + v;
            const size_t off = ((size_t)(b * NK + (k0 + ln)) * NC + i) * 2;
            float2 val;
            val.x = s5 * xre[v];
            val.y = s5 * xim[v];
            *reinterpret_cast<float2*>(out + off) = val;
        }
        // next iteration's top barrier orders T reuse and buffer swap
    }
}

extern "C" void kernel_launch(void* const* d_in, const int* in_sizes, int n_in,
                              void* d_out, int out_size, void* d_ws, size_t ws_size,
                              hipStream_t stream) {
    (void)in_sizes; (void)n_in; (void)out_size; (void)d_ws; (void)ws_size;
    const float* u_re = (const float*)d_in[0];   // (256,2,8,8)
    const float* u_im = (const float*)d_in[1];
    const float* b_re = (const float*)d_in[2];   // (512,8,8,2)
    const float* b_im = (const float*)d_in[3];
    float* out = (float*)d_out;                  // (256,512,128) complex64 interleaved

    const size_t shmem = (size_t)(2 * NC * DSTR + 6 * KTILE * DSTR) * sizeof(float);
    dd_wilson_wmma<<<dim3(NB), dim3(256), shmem, stream>>>(u_re, u_im, b_re, b_im, out);
}